// AttnPooling_60816736911837
// MI455X (gfx1250) — compile-verified
//
#include <hip/hip_runtime.h>
#include <math.h>

// ---------------- problem constants (match reference) ----------------
#define B_G    64
#define N_PER  2048
#define N_TOT  (B_G * N_PER)      // 131072 nodes
#define E_TOT  4194304            // edges
#define C_IN   256                // channels
#define K_KEEP 1024               // kept per graph (ratio 0.5)
#define KEPT   (B_G * K_KEEP)     // 65536 kept nodes

typedef __attribute__((ext_vector_type(2))) float v2f;
typedef __attribute__((ext_vector_type(8))) float v8f;

// ---------------------------------------------------------------------
// Kernel 0: node_idx = -1 over all N nodes (workspace init; deterministic)
// ---------------------------------------------------------------------
__global__ __launch_bounds__(256) void init_nodeidx_kernel(int* __restrict__ nidx) {
    int i = blockIdx.x * 256 + threadIdx.x;
    if (i < N_TOT) nidx[i] = -1;
}

// ---------------------------------------------------------------------
// Kernel 1: score[n] = tanh(dot(x[n,:], q)) via V_WMMA_F32_16X16X4_F32.
// One wave computes 16 node scores. A-frag (16x4 f32): lane<16 holds row
// M=lane, K={kb,kb+1}; lane>=16 holds K={kb+2,kb+3}. B-frag (4x16): the
// query chunk broadcast across all 16 columns -> every column of D holds
// the dot product. C/D layout: lane half h, VGPR v -> row v + 8h.
// ---------------------------------------------------------------------
__global__ __launch_bounds__(256)
void score_wmma_kernel(const float* __restrict__ x,
                       const float* __restrict__ q,
                       float* __restrict__ score) {
    __shared__ float qs[C_IN];
    const int tid = threadIdx.x;
    qs[tid] = q[tid];                    // blockDim == C_IN == 256
    __syncthreads();

    const int wave = tid >> 5;
    const int lane = tid & 31;
    const int half = lane >> 4;          // 0: K pair {0,1}; 1: K pair {2,3}
    const int row  = lane & 15;          // M index for the A fragment

    const long node_base = ((long)blockIdx.x * 8 + wave) * 16;
    const float* xr = x + (node_base + row) * C_IN + 2 * half;

    v8f acc = {};
#pragma unroll 4
    for (int kb = 0; kb < C_IN; kb += 4) {
        v2f a = *(const v2f*)(xr + kb);          // global_load_b64, 8B aligned
        v2f b;
        b.x = qs[kb + 2 * half];                 // ds_load: q chunk, bcast over N
        b.y = qs[kb + 2 * half + 1];
        acc = __builtin_amdgcn_wmma_f32_16x16x4_f32(
            /*neg_a=*/false, a, /*neg_b=*/false, b,
            /*c_mod=*/(short)0, acc, /*reuse_a=*/false, /*reuse_b=*/false);
    }

    if (row == 0) {                              // lanes 0 and 16 own all 16 rows
#pragma unroll
        for (int v = 0; v < 8; ++v)
            score[node_base + half * 8 + v] = tanhf(acc[v]);
    }
}

// ---------------------------------------------------------------------
// Kernel 2: per-graph top-K via full bitonic sort of 2048 (score, idx)
// pairs in LDS. Order matches lax.top_k: score descending, ties by
// ascending index. Writes perm / score_perm / batch directly to d_out
// regions and scatters node_idx[perm[j]] = global kept rank.
// ---------------------------------------------------------------------
__global__ __launch_bounds__(1024)
void topk_sort_kernel(const float* __restrict__ score,
                      int* __restrict__ o_perm,
                      float* __restrict__ o_scoreperm,
                      int* __restrict__ o_batch,
                      int* __restrict__ nidx) {
    __shared__ float key[N_PER];
    __shared__ int   val[N_PER];
    const int b = blockIdx.x;
    const int t = threadIdx.x;

    for (int i = t; i < N_PER; i += 1024) {
        key[i] = score[b * N_PER + i];
        val[i] = i;
    }
    __syncthreads();

    for (int k = 2; k <= N_PER; k <<= 1) {
        for (int j = k >> 1; j > 0; j >>= 1) {
            for (int i = t; i < N_PER; i += 1024) {
                int l = i ^ j;
                if (l > i) {
                    bool dir = ((i & k) == 0);   // "better-first" region
                    float ki = key[i], kl = key[l];
                    int   vi = val[i], vl = val[l];
                    bool l_better = (kl > ki) || (kl == ki && vl < vi);
                    bool i_better = (ki > kl) || (ki == kl && vi < vl);
                    if (dir ? l_better : i_better) {
                        key[i] = kl; key[l] = ki;
                        val[i] = vl; val[l] = vi;
                    }
                }
            }
            __syncthreads();
        }
    }

    if (t < K_KEEP) {
        int p   = b * N_PER + val[t];            // global node id kept at rank t
        int out = b * K_KEEP + t;                // new (relabeled) id
        o_perm[out]      = p;
        o_scoreperm[out] = key[t];
        o_batch[out]     = b;
        nidx[p]          = out;                  // scatter for edge relabel
    }
}

// ---------------------------------------------------------------------
// Kernel 3: x_out[r, :] = x[perm[r], :] * score_perm[r]. float4 rows;
// 4 rows per 256-thread block, 64 lanes per row.
// ---------------------------------------------------------------------
__global__ __launch_bounds__(256)
void gather_scale_kernel(const float* __restrict__ x,
                         const int* __restrict__ perm,
                         const float* __restrict__ sp,
                         float* __restrict__ xout) {
    const int tid = threadIdx.x;
    const int row = blockIdx.x * 4 + (tid >> 6);
    const int c4  = tid & 63;                    // C/4 == 64 float4 per row
    const int p   = perm[row];
    const float s = sp[row];
    float4 v = ((const float4*)(x + (long)p * C_IN))[c4];   // global_load_b128
    v.x *= s; v.y *= s; v.z *= s; v.w *= s;
    ((float4*)(xout + (long)row * C_IN))[c4] = v;
}

// ---------------------------------------------------------------------
// Kernel 4: relabel edges through node_idx, mask edges touching dropped
// nodes with -1 (both rows). 4 edges per thread, int4 vectorized.
// ---------------------------------------------------------------------
__global__ __launch_bounds__(256)
void edge_relabel_kernel(const int* __restrict__ ei,
                         const int* __restrict__ nidx,
                         int* __restrict__ o_ei) {
    const long e4 = (long)blockIdx.x * 256 + threadIdx.x;   // 4 edges at 4*e4
    int4 s = ((const int4*)ei)[e4];
    int4 d = ((const int4*)(ei + E_TOT))[e4];

    int rs[4] = { nidx[s.x], nidx[s.y], nidx[s.z], nidx[s.w] };
    int rd[4] = { nidx[d.x], nidx[d.y], nidx[d.z], nidx[d.w] };
#pragma unroll
    for (int c = 0; c < 4; ++c) {
        bool ok = (rs[c] >= 0) && (rd[c] >= 0);
        if (!ok) { rs[c] = -1; rd[c] = -1; }
    }
    int4 os = { rs[0], rs[1], rs[2], rs[3] };
    int4 od = { rd[0], rd[1], rd[2], rd[3] };
    ((int4*)o_ei)[e4]                 = os;      // row 0
    ((int4*)(o_ei + E_TOT))[e4]       = od;      // row 1
}

// ---------------------------------------------------------------------
// Host-side launch (graph-capture safe: only kernel launches on stream).
// d_in order: x [N*C] f32, edge_index [2*E] i32, query_vec [C] f32,
//             batch [N] i32 (unused; batch_out derived from graph id).
// d_out (flat, reference return order):
//   x_out [KEPT*C] f32 | new_ei [2*E] i32 | batch_out [KEPT] i32 |
//   perm [KEPT] i32 | score_perm [KEPT] f32
// ---------------------------------------------------------------------
extern "C" void kernel_launch(void* const* d_in, const int* in_sizes, int n_in,
                              void* d_out, int out_size, void* d_ws, size_t ws_size,
                              hipStream_t stream) {
    const float* x  = (const float*)d_in[0];
    const int*   ei = (const int*)d_in[1];
    const float* q  = (const float*)d_in[2];

    float* outf      = (float*)d_out;
    float* o_x       = outf;
    int*   o_ei      = (int*)  (outf + (long)KEPT * C_IN);
    int*   o_batch   = (int*)  (outf + (long)KEPT * C_IN + 2L * E_TOT);
    int*   o_perm    = o_batch + KEPT;
    float* o_sp      = (float*)(o_perm + KEPT);

    float* score = (float*)d_ws;                               // N floats
    int*   nidx  = (int*)((char*)d_ws + (size_t)N_TOT * 4);    // N ints

    init_nodeidx_kernel<<<N_TOT / 256, 256, 0, stream>>>(nidx);
    score_wmma_kernel<<<N_TOT / 128, 256, 0, stream>>>(x, q, score);
    topk_sort_kernel<<<B_G, 1024, 0, stream>>>(score, o_perm, o_sp, o_batch, nidx);
    gather_scale_kernel<<<KEPT / 4, 256, 0, stream>>>(x, o_perm, o_sp, o_x);
    edge_relabel_kernel<<<E_TOT / 4 / 256, 256, 0, stream>>>(ei, nidx, o_ei);
}